// SqrllGate_10900626997644
// MI455X (gfx1250) — compile-verified
//
#include <hip/hip_runtime.h>

#define N_IN   1024
#define N_MEMD 2048
#define N_OUTP 1024
#define BATCH  4
#define SEQT   4096
#define MROWS  (BATCH * SEQT)      // 16384
#define CHUNKS 16
#define CHUNK_T (SEQT / CHUNKS)    // 256

typedef __attribute__((ext_vector_type(2))) float v2f;
typedef __attribute__((ext_vector_type(8))) float v8f;

__device__ __forceinline__ float fsig(float v) {
    return 1.0f / (1.0f + __expf(-v));
}

__device__ __forceinline__ v8f wmma_f32_k4(v2f a, v2f b, v8f c) {
    // D = A(16x4,f32) * B(4x16,f32) + C(16x16,f32)
    return __builtin_amdgcn_wmma_f32_16x16x4_f32(
        /*neg_a=*/false, a, /*neg_b=*/false, b,
        /*c_mod=*/(short)0, c, /*reuse_a=*/false, /*reuse_b=*/false);
}

// ---------------------------------------------------------------------------
// Kernel 1: fused four-gate GEMM.
//   For every row of x (B*T rows, K=N_IN) compute projections against
//   wi, wig, wf, wog (N=N_MEMD), apply biases + sigmoids, emit
//   y = (wi.x)*sig(wig.x+b), r = sig(wf.x+b), g = sig(wog.x+b).
// Wave tile: 64 rows x 16 cols, 4 gates -> 16 f32 accumulators.
// Block: 8 waves -> 64 rows x 128 cols per workgroup.
// ---------------------------------------------------------------------------
__global__ __launch_bounds__(256) void k_gates(
    const float* __restrict__ x,
    const float* __restrict__ wi_w,  const float* __restrict__ wig_w,
    const float* __restrict__ wf_w,  const float* __restrict__ wog_w,
    const float* __restrict__ wig_b, const float* __restrict__ wf_b,
    const float* __restrict__ wog_b,
    float* __restrict__ yb, float* __restrict__ rb, float* __restrict__ gb)
{
    const int lane = threadIdx.x & 31;
    const int wave = threadIdx.x >> 5;
    const int lr   = lane & 15;   // row-within-tile / col-within-tile selector
    const int lk   = lane >> 4;   // K half selector
    const int n0   = blockIdx.x * 128 + wave * 16;
    const int m0   = blockIdx.y * 64;

    v8f acc[4][4];
#pragma unroll
    for (int g = 0; g < 4; ++g)
#pragma unroll
        for (int mt = 0; mt < 4; ++mt)
            acc[g][mt] = (v8f){0.f, 0.f, 0.f, 0.f, 0.f, 0.f, 0.f, 0.f};

    const float* arow[4];
#pragma unroll
    for (int mt = 0; mt < 4; ++mt)
        arow[mt] = x + (size_t)(m0 + mt * 16 + lr) * N_IN + 2 * lk;

    const float* brow[4];
    brow[0] = wi_w  + (size_t)(n0 + lr) * N_IN + 2 * lk;
    brow[1] = wig_w + (size_t)(n0 + lr) * N_IN + 2 * lk;
    brow[2] = wf_w  + (size_t)(n0 + lr) * N_IN + 2 * lk;
    brow[3] = wog_w + (size_t)(n0 + lr) * N_IN + 2 * lk;

#pragma unroll 2
    for (int k0 = 0; k0 < N_IN; k0 += 4) {
        v2f a[4];
#pragma unroll
        for (int mt = 0; mt < 4; ++mt)
            a[mt] = *(const v2f*)(arow[mt] + k0);
#pragma unroll
        for (int g = 0; g < 4; ++g) {
            v2f bf = *(const v2f*)(brow[g] + k0);
#pragma unroll
            for (int mt = 0; mt < 4; ++mt)
                acc[g][mt] = wmma_f32_k4(a[mt], bf, acc[g][mt]);
        }
    }

    const int n = n0 + lr;
    const float bwig = wig_b[n];
    const float bwf  = wf_b[n];
    const float bwog = wog_b[n];

#pragma unroll
    for (int mt = 0; mt < 4; ++mt) {
#pragma unroll
        for (int v = 0; v < 8; ++v) {
            const int row = m0 + mt * 16 + v + 8 * lk;   // C layout: M = v (+8 for lanes 16-31)
            const size_t o = (size_t)row * N_MEMD + n;
            const float ai  = acc[0][mt][v];
            const float aig = acc[1][mt][v] + bwig;
            const float af  = acc[2][mt][v] + bwf;
            const float aog = acc[3][mt][v] + bwog;
            yb[o] = ai * fsig(aig);
            rb[o] = fsig(af);
            gb[o] = fsig(aog);
        }
    }
}

// ---------------------------------------------------------------------------
// Kernel 2a: local (per-chunk) scan. Overwrites y with local running carry,
// r with running prefix product of r within the chunk.
// One thread per (chunk, b, n): 131072 threads.
// ---------------------------------------------------------------------------
__global__ __launch_bounds__(256) void k_scan_local(
    float* __restrict__ yb, float* __restrict__ rb)
{
    const int idx = blockIdx.x * blockDim.x + threadIdx.x;
    const int n    = idx & (N_MEMD - 1);
    const int rest = idx >> 11;
    const int b    = rest & (BATCH - 1);
    const int c    = rest >> 2;

    size_t off = (size_t)(b * SEQT + c * CHUNK_T) * N_MEMD + n;
    float carry = 0.0f;
    float prod  = 1.0f;
#pragma unroll 4
    for (int i = 0; i < CHUNK_T; ++i) {
        const float yv = yb[off];
        const float rv = rb[off];
        carry = fmaf(rv, carry, yv);
        prod *= rv;
        yb[off] = carry;
        rb[off] = prod;
        off += N_MEMD;
    }
}

// ---------------------------------------------------------------------------
// Kernel 2b: combine chunk carries sequentially (16 chunks), seeded with mem.
// Emits per-chunk incoming carry and mem_new (= scanned y at t = T-1).
// One thread per (b, n): 8192 threads.
// ---------------------------------------------------------------------------
__global__ __launch_bounds__(256) void k_scan_carry(
    const float* __restrict__ yb, const float* __restrict__ rb,
    const float* __restrict__ mem,
    float* __restrict__ incb, float* __restrict__ out)
{
    const int idx = blockIdx.x * blockDim.x + threadIdx.x;
    const int n = idx & (N_MEMD - 1);
    const int b = idx >> 11;

    float inc = mem[(size_t)b * N_MEMD + n];
#pragma unroll
    for (int c = 0; c < CHUNKS; ++c) {
        incb[((size_t)c * BATCH + b) * N_MEMD + n] = inc;
        const size_t last =
            (size_t)(b * SEQT + c * CHUNK_T + CHUNK_T - 1) * N_MEMD + n;
        inc = fmaf(rb[last], inc, yb[last]);   // y_end + P_end * inc
    }
    // mem_new lives after the main output tensor
    out[(size_t)MROWS * N_OUTP + (size_t)b * N_MEMD + n] = inc;
}

// ---------------------------------------------------------------------------
// Kernel 2c: apply incoming carry (y'_t = y_t + P_t * inc), then fuse
// softsign + output gate:  h = y'/(1+|y'|) * g.  h overwrites y.
// One thread per (chunk, b, n).
// ---------------------------------------------------------------------------
__global__ __launch_bounds__(256) void k_scan_apply(
    float* __restrict__ yb, const float* __restrict__ rb,
    const float* __restrict__ gb, const float* __restrict__ incb)
{
    const int idx = blockIdx.x * blockDim.x + threadIdx.x;
    const int n    = idx & (N_MEMD - 1);
    const int rest = idx >> 11;
    const int b    = rest & (BATCH - 1);
    const int c    = rest >> 2;

    const float inc = incb[((size_t)c * BATCH + b) * N_MEMD + n];
    size_t off = (size_t)(b * SEQT + c * CHUNK_T) * N_MEMD + n;
#pragma unroll 4
    for (int i = 0; i < CHUNK_T; ++i) {
        const float yv = fmaf(rb[off], inc, yb[off]);
        const float ss = yv / (1.0f + fabsf(yv));
        yb[off] = ss * gb[off];
        off += N_MEMD;
    }
}

// ---------------------------------------------------------------------------
// Kernel 3: output projection  out = h (MROWS x N_MEMD) . wo_w^T (N_MEMD x N_OUTP)
// Wave tile: 64 rows x 32 cols -> 8 accumulators. Block: 8 waves.
// ---------------------------------------------------------------------------
__global__ __launch_bounds__(256) void k_out(
    const float* __restrict__ h, const float* __restrict__ wo_w,
    float* __restrict__ out)
{
    const int lane = threadIdx.x & 31;
    const int wave = threadIdx.x >> 5;
    const int lr   = lane & 15;
    const int lk   = lane >> 4;
    const int n0   = blockIdx.x * 256 + wave * 32;
    const int m0   = blockIdx.y * 64;

    v8f acc[2][4];
#pragma unroll
    for (int nt = 0; nt < 2; ++nt)
#pragma unroll
        for (int mt = 0; mt < 4; ++mt)
            acc[nt][mt] = (v8f){0.f, 0.f, 0.f, 0.f, 0.f, 0.f, 0.f, 0.f};

    const float* arow[4];
#pragma unroll
    for (int mt = 0; mt < 4; ++mt)
        arow[mt] = h + (size_t)(m0 + mt * 16 + lr) * N_MEMD + 2 * lk;
    const float* brow[2];
#pragma unroll
    for (int nt = 0; nt < 2; ++nt)
        brow[nt] = wo_w + (size_t)(n0 + nt * 16 + lr) * N_MEMD + 2 * lk;

#pragma unroll 2
    for (int k0 = 0; k0 < N_MEMD; k0 += 4) {
        v2f a[4];
#pragma unroll
        for (int mt = 0; mt < 4; ++mt)
            a[mt] = *(const v2f*)(arow[mt] + k0);
#pragma unroll
        for (int nt = 0; nt < 2; ++nt) {
            v2f bf = *(const v2f*)(brow[nt] + k0);
#pragma unroll
            for (int mt = 0; mt < 4; ++mt)
                acc[nt][mt] = wmma_f32_k4(a[mt], bf, acc[nt][mt]);
        }
    }

#pragma unroll
    for (int nt = 0; nt < 2; ++nt) {
        const int n = n0 + nt * 16 + lr;
#pragma unroll
        for (int mt = 0; mt < 4; ++mt) {
#pragma unroll
            for (int v = 0; v < 8; ++v) {
                const int row = m0 + mt * 16 + v + 8 * lk;
                out[(size_t)row * N_OUTP + n] = acc[nt][mt][v];
            }
        }
    }
}

extern "C" void kernel_launch(void* const* d_in, const int* in_sizes, int n_in,
                              void* d_out, int out_size, void* d_ws, size_t ws_size,
                              hipStream_t stream) {
    (void)in_sizes; (void)n_in; (void)out_size; (void)ws_size;
    const float* x     = (const float*)d_in[0];
    const float* mem   = (const float*)d_in[1];
    const float* wf_w  = (const float*)d_in[2];
    const float* wf_b  = (const float*)d_in[3];
    const float* wi_w  = (const float*)d_in[4];
    const float* wig_w = (const float*)d_in[5];
    const float* wig_b = (const float*)d_in[6];
    const float* wog_w = (const float*)d_in[7];
    const float* wog_b = (const float*)d_in[8];
    const float* wo_w  = (const float*)d_in[9];
    float* out = (float*)d_out;

    // Workspace layout: y | r | g | chunk carries
    const size_t HM = (size_t)MROWS * N_MEMD;          // 33,554,432 floats each
    float* ws   = (float*)d_ws;
    float* yb   = ws;
    float* rb   = ws + HM;
    float* gb   = ws + 2 * HM;
    float* incb = ws + 3 * HM;                          // CHUNKS*BATCH*N_MEMD floats

    // 1) fused gate GEMMs + activations
    k_gates<<<dim3(N_MEMD / 128, MROWS / 64), 256, 0, stream>>>(
        x, wi_w, wig_w, wf_w, wog_w, wig_b, wf_b, wog_b, yb, rb, gb);

    // 2) chunked linear-recurrence scan
    k_scan_local<<<(BATCH * N_MEMD * CHUNKS) / 256, 256, 0, stream>>>(yb, rb);
    k_scan_carry<<<(BATCH * N_MEMD) / 256, 256, 0, stream>>>(yb, rb, mem, incb, out);
    k_scan_apply<<<(BATCH * N_MEMD * CHUNKS) / 256, 256, 0, stream>>>(yb, rb, gb, incb);

    // 3) output projection
    k_out<<<dim3(N_OUTP / 256, MROWS / 64), 256, 0, stream>>>(yb, wo_w, out);
}